// molGCN_only_node_41618233099022
// MI455X (gfx1250) — compile-verified
//
#include <hip/hip_runtime.h>

// ---------------------------------------------------------------------------
// molGCN on MI455X (gfx1250, wave32).
// - GEMMs: V_WMMA_F32_16X16X4_F32, one wave per 16x64 row-strip (4
//   accumulators) so each A fragment feeds 4 WMMAs.
// - Sparse phases: L2-resident (12.8 MB node features << 192 MB L2);
//   float4 (b128) gathers, f32 global atomics for the scatter-add.
// ---------------------------------------------------------------------------

typedef __attribute__((ext_vector_type(2))) float v2f;
typedef __attribute__((ext_vector_type(8))) float v8f;

#define NNODES   50000
#define NEDGES   800000
#define IN_CH    128
#define HID      64
#define OUT_CH   64
#define NCLASSES 10
#define NGRAPHS  256

__global__ void zero4_kernel(float4* __restrict__ p, int n4) {
  int i = blockIdx.x * blockDim.x + threadIdx.x;
  if (i < n4) p[i] = make_float4(0.f, 0.f, 0.f, 0.f);
}

__global__ void degree_kernel(const int* __restrict__ dst, float* __restrict__ deg) {
  int e = blockIdx.x * blockDim.x + threadIdx.x;
  if (e < NEDGES) atomicAdd(&deg[dst[e]], 1.0f);
}

__global__ void dinv_kernel(const float* __restrict__ deg, float* __restrict__ dinv) {
  int i = blockIdx.x * blockDim.x + threadIdx.x;
  if (i < NNODES) dinv[i] = rsqrtf(deg[i] + 1.0f);
}

// C[M x 64] = A[M x K] * W[K x 64].  M multiple of 16, K multiple of 4.
// One wave32 per 16-row strip; 4 accumulator tiles cover all 64 columns so
// each A fragment is reused by 4 consecutive WMMAs.
// Lane layouts per CDNA5 ISA 7.12.2 (32-bit operands):
//   A 16x4: lane = M (mod 16); vgpr0={K=0|K=2}, vgpr1={K=1|K=3} (lane halves)
//   B 4x16: lane = N (mod 16); same K split across lane halves
//   C/D:    vgpr v, lane l -> M = v + 8*(l>>4), N = l&15
__global__ void wmma_gemm_f32(const float* __restrict__ A, const float* __restrict__ W,
                              float* __restrict__ C, int M, int K) {
  const int lane = threadIdx.x & 31;
  const int wave = threadIdx.x >> 5;
  const int mt = blockIdx.x * (blockDim.x >> 5) + wave;
  if (mt >= (M >> 4)) return;   // wave-uniform: EXEC stays all-1s for WMMA

  const int half = lane >> 4;   // 0: lanes 0-15, 1: lanes 16-31 (upper K pair)
  const int sub  = lane & 15;
  const int arow = mt * 16 + sub;

  v8f acc[4];
  acc[0] = (v8f){};
  acc[1] = (v8f){};
  acc[2] = (v8f){};
  acc[3] = (v8f){};

  for (int k0 = 0; k0 < K; k0 += 4) {
    const int ka = k0 + 2 * half;
    v2f a;
    a.x = A[arow * K + ka];
    a.y = A[arow * K + ka + 1];
#pragma unroll
    for (int nt = 0; nt < 4; ++nt) {
      v2f b;
      b.x = W[ka * 64 + nt * 16 + sub];
      b.y = W[(ka + 1) * 64 + nt * 16 + sub];
      acc[nt] = __builtin_amdgcn_wmma_f32_16x16x4_f32(
          /*neg_a=*/false, a, /*neg_b=*/false, b,
          /*c_mod=*/(short)0, acc[nt], /*reuse_a=*/false, /*reuse_b=*/false);
    }
  }

#pragma unroll
  for (int nt = 0; nt < 4; ++nt) {
#pragma unroll
    for (int v = 0; v < 8; ++v) {
      const int m = mt * 16 + v + 8 * half;
      C[m * 64 + nt * 16 + sub] = acc[nt][v];
    }
  }
}

// One thread per (edge, channel-quad): b128 gather + 4 f32 atomic scatters.
__global__ void msg_kernel(const float* __restrict__ h, const int* __restrict__ src,
                           const int* __restrict__ dst, const float* __restrict__ dinv,
                           float* __restrict__ agg) {
  long long tid = (long long)blockIdx.x * blockDim.x + threadIdx.x;
  if (tid >= (long long)NEDGES * 16) return;
  const int e = (int)(tid >> 4);
  const int q = (int)(tid & 15);
  const int s = src[e];
  const int d = dst[e];
  const float norm = dinv[s] * dinv[d];
  const float4 hv = *(const float4*)(h + s * 64 + q * 4);
  float* ap = agg + d * 64 + q * 4;
  atomicAdd(ap + 0, hv.x * norm);
  atomicAdd(ap + 1, hv.y * norm);
  atomicAdd(ap + 2, hv.z * norm);
  atomicAdd(ap + 3, hv.w * norm);
}

// In place: agg = (relu?)(agg + h*dinv^2 + bias)   (float4 per thread)
__global__ void finalize_kernel(float* __restrict__ agg, const float* __restrict__ h,
                                const float* __restrict__ dinv, const float* __restrict__ bias,
                                int do_relu) {
  long long tid = (long long)blockIdx.x * blockDim.x + threadIdx.x;
  if (tid >= (long long)NNODES * 16) return;
  const int i = (int)(tid >> 4);
  const int q = (int)(tid & 15);
  const float di = dinv[i];
  const float d2 = di * di;
  const float4 av = *(const float4*)(agg + i * 64 + q * 4);
  const float4 hv = *(const float4*)(h + i * 64 + q * 4);
  const float4 bv = *(const float4*)(bias + q * 4);
  float4 r;
  r.x = av.x + hv.x * d2 + bv.x;
  r.y = av.y + hv.y * d2 + bv.y;
  r.z = av.z + hv.z * d2 + bv.z;
  r.w = av.w + hv.w * d2 + bv.w;
  if (do_relu) {
    r.x = fmaxf(r.x, 0.f); r.y = fmaxf(r.y, 0.f);
    r.z = fmaxf(r.z, 0.f); r.w = fmaxf(r.w, 0.f);
  }
  *(float4*)(agg + i * 64 + q * 4) = r;
}

__global__ void pool_kernel(const float* __restrict__ h, const int* __restrict__ batch,
                            float* __restrict__ pooled, float* __restrict__ counts) {
  long long tid = (long long)blockIdx.x * blockDim.x + threadIdx.x;
  if (tid >= (long long)NNODES * 16) return;
  const int i = (int)(tid >> 4);
  const int q = (int)(tid & 15);
  const int g = batch[i];
  const float4 hv = *(const float4*)(h + i * 64 + q * 4);
  float* pp = pooled + g * 64 + q * 4;
  atomicAdd(pp + 0, hv.x);
  atomicAdd(pp + 1, hv.y);
  atomicAdd(pp + 2, hv.z);
  atomicAdd(pp + 3, hv.w);
  if (q == 0) atomicAdd(&counts[g], 1.0f);
}

__global__ void classify_kernel(const float* __restrict__ pooled, const float* __restrict__ counts,
                                const float* __restrict__ Wl, const float* __restrict__ bl,
                                float* __restrict__ out) {
  int tid = blockIdx.x * blockDim.x + threadIdx.x;
  if (tid >= NGRAPHS * NCLASSES) return;
  const int g = tid / NCLASSES;
  const int k = tid - g * NCLASSES;
  const float inv = 1.0f / fmaxf(counts[g], 1.0f);
  float s = 0.0f;
#pragma unroll
  for (int j = 0; j < OUT_CH; ++j) s += pooled[g * OUT_CH + j] * Wl[j * NCLASSES + k];
  out[tid] = s * inv + bl[k];
}

static inline int nblk(long long n, int tb) { return (int)((n + tb - 1) / tb); }

extern "C" void kernel_launch(void* const* d_in, const int* in_sizes, int n_in,
                              void* d_out, int out_size, void* d_ws, size_t ws_size,
                              hipStream_t stream) {
  (void)in_sizes; (void)n_in; (void)out_size; (void)ws_size;

  const float* x   = (const float*)d_in[0];
  const int*   ei  = (const int*)d_in[1];     // shape (2, E) row-major
  const int*   src = ei;
  const int*   dst = ei + NEDGES;
  // d_in[2] = edge_attr (unused by reference math)
  const int*   batch = (const int*)d_in[3];
  const float* W1 = (const float*)d_in[4];
  const float* b1 = (const float*)d_in[5];
  const float* W2 = (const float*)d_in[6];
  const float* b2 = (const float*)d_in[7];
  const float* Wl = (const float*)d_in[8];
  const float* bl = (const float*)d_in[9];
  float* out = (float*)d_out;

  // Workspace layout (floats): ~26 MB; every sub-buffer 16B-aligned.
  float* ws     = (float*)d_ws;
  float* deg    = ws;  ws += NNODES;
  float* dinv   = ws;  ws += NNODES;
  float* A      = ws;  ws += (size_t)NNODES * 64;  // h1, later h2
  float* B      = ws;  ws += (size_t)NNODES * 64;  // agg1/h1out, later agg2/h2out
  float* pooled = ws;  ws += NGRAPHS * 64;
  float* counts = ws;  ws += NGRAPHS;

  const int TB = 256;

  // degrees + symmetric normalization
  zero4_kernel<<<nblk(NNODES / 4, TB), TB, 0, stream>>>((float4*)deg, NNODES / 4);
  degree_kernel<<<nblk(NEDGES, TB), TB, 0, stream>>>(dst, deg);
  dinv_kernel<<<nblk(NNODES, TB), TB, 0, stream>>>(deg, dinv);

  // ---- layer 1: h1 = x @ W1 ; aggregate ; relu ----
  wmma_gemm_f32<<<nblk((long long)(NNODES / 16) * 32, TB), TB, 0, stream>>>(
      x, W1, A, NNODES, IN_CH);
  zero4_kernel<<<nblk((long long)NNODES * 16, TB), TB, 0, stream>>>((float4*)B, NNODES * 16);
  msg_kernel<<<nblk((long long)NEDGES * 16, TB), TB, 0, stream>>>(A, src, dst, dinv, B);
  finalize_kernel<<<nblk((long long)NNODES * 16, TB), TB, 0, stream>>>(B, A, dinv, b1, 1);

  // ---- layer 2: h2 = h1out @ W2 ; aggregate (no relu) ----
  wmma_gemm_f32<<<nblk((long long)(NNODES / 16) * 32, TB), TB, 0, stream>>>(
      B, W2, A, NNODES, HID);
  zero4_kernel<<<nblk((long long)NNODES * 16, TB), TB, 0, stream>>>((float4*)B, NNODES * 16);
  msg_kernel<<<nblk((long long)NEDGES * 16, TB), TB, 0, stream>>>(A, src, dst, dinv, B);
  finalize_kernel<<<nblk((long long)NNODES * 16, TB), TB, 0, stream>>>(B, A, dinv, b2, 0);

  // ---- global mean pool + linear head ----
  zero4_kernel<<<nblk((NGRAPHS * 64 + NGRAPHS) / 4, TB), TB, 0, stream>>>(
      (float4*)pooled, (NGRAPHS * 64 + NGRAPHS) / 4);
  pool_kernel<<<nblk((long long)NNODES * 16, TB), TB, 0, stream>>>(B, batch, pooled, counts);
  classify_kernel<<<nblk(NGRAPHS * NCLASSES, TB), TB, 0, stream>>>(pooled, counts, Wl, bl, out);
}